// SKA_38431367364735
// MI455X (gfx1250) — compile-verified
//
#include <hip/hip_runtime.h>
#include <stdint.h>

namespace {
constexpr int kN   = 8;
constexpr int kIC  = 512;
constexpr int kH   = 96;
constexpr int kW   = 96;
constexpr int kWC  = 8;
constexpr int kKS  = 7;
constexpr int kPad = 3;
constexpr int kG   = kIC / kWC;          // 64 channel groups sharing a weight channel
constexpr int kTH  = 8;                  // output rows per block tile
constexpr int kNT  = 192;                // threads per block (24 col-groups x 8 rows)
constexpr int kXRows = kTH + kKS - 1;    // 14 halo rows
constexpr int kXStr  = 104;              // padded LDS x-row stride in floats
constexpr int kWTile = kKS * kKS * kTH * kW;  // 37632 floats = 150528 B
constexpr int kXTile = kXRows * kXStr;        // 1456 floats  = 5824 B (x2 buffers)
constexpr int kHW  = kH * kW;            // 9216
}

typedef uint32_t u32x4 __attribute__((ext_vector_type(4)));
typedef uint32_t u32x8 __attribute__((ext_vector_type(8)));

__global__ __launch_bounds__(kNT)
void ska_forward_kernel(const float* __restrict__ x,
                        const float* __restrict__ w,
                        float* __restrict__ out) {
  __shared__ __align__(16) float WT[kWTile];      // weight tile: [tap][row(8)][col(96)]
  __shared__ __align__(16) float XT[2][kXTile];   // double-buffered x halo tile

  const int tid    = threadIdx.x;
  const int ytile  = blockIdx.x;          // 0..11
  const int nk     = blockIdx.y;          // 0..63  (nn*8 + k)
  const int nn     = nk >> 3;
  const int k      = nk & 7;              // weight channel
  const int y0     = ytile * kTH;
  const int rowIdx = tid / 24;            // 0..7  (output row within tile)
  const int x0     = (tid % 24) * 4;      // 0..92 (4-wide output strip)

  // ---- Tensor Data Mover: DMA the whole weight tile into LDS with one D# ----
  // 2D tile: dim0 = 768 contiguous floats (8 rows x 96 cols of one tap),
  //          dim1 = 49 taps, row stride 9216 floats; LDS dest packs contiguously.
  if (tid < 32) {   // wave 0 issues the TDM op (EXEC ignored; once per wave)
    const uint64_t gaddr = (uint64_t)(uintptr_t)(
        w + (size_t)(nn * kWC + k) * (kKS * kKS) * kHW + (size_t)y0 * kW);
    const uint32_t ldsw = (uint32_t)(uintptr_t)(&WT[0]);

    u32x4 g0;
    g0[0] = 1u;                                        // count=1, user descriptor
    g0[1] = ldsw;                                      // lds_addr (bytes)
    g0[2] = (uint32_t)(gaddr & 0xFFFFFFFFu);           // global_addr[31:0]
    g0[3] = (uint32_t)((gaddr >> 32) & 0x01FFFFFFu)    // global_addr[56:32]
            | 0x80000000u;                             // type=2 ("image") at [127:126]

    u32x8 g1;
    g1[0] = 0x00020000u;                               // workgroup_mask=0, data_size=2 (4B)
    g1[1] = (uint32_t)(kHW & 0xFFFF) << 16;            // tensor_dim0[15:0] = 9216
    g1[2] = ((uint32_t)(kKS * kKS)) << 16;             // tensor_dim0[31:16]=0 | tensor_dim1=49
    g1[3] = ((uint32_t)(kTH * kW)) << 16;              // tensor_dim1 hi=0 | tile_dim0=768
    g1[4] = (uint32_t)(kKS * kKS);                     // tile_dim1=49 | tile_dim2=0
    g1[5] = (uint32_t)kHW;                             // tensor_dim0_stride[31:0] = 9216
    g1[6] = 0u;                                        // stride0 hi | tensor_dim1_stride lo
    g1[7] = 0u;                                        // tensor_dim1_stride hi

    u32x4 g2 = {0u, 0u, 0u, 0u};                       // dims 2/3 unused (tile_dim=0)
    u32x4 g3 = {0u, 0u, 0u, 0u};                       // dim 4 unused

    asm volatile("tensor_load_to_lds %0, %1, %2, %3"
                 :: "s"(g0), "s"(g1), "s"(g2), "s"(g3) : "memory");
  }

  // ---- zero both x buffers once (covers halo cells never written by async loads) ----
  for (int i = tid; i < 2 * kXTile; i += kNT) (&XT[0][0])[i] = 0.0f;
  asm volatile("s_wait_dscnt 0x0" ::: "memory");
  __syncthreads();

  // ---- precompute per-thread x-halo copy slots (7 b32 async issues per thread) ----
  const uint32_t ldsX0 = (uint32_t)(uintptr_t)(&XT[0][0]);
  const uint32_t ldsX1 = (uint32_t)(uintptr_t)(&XT[1][0]);
  uint32_t xgoff[7], xloff[7];
  bool xok[7];
#pragma unroll
  for (int it = 0; it < 7; ++it) {
    const int idx = tid + kNT * it;       // 0..1343 covers 14x96 halo cells
    const int p   = idx / kW;             // padded tile row 0..13
    const int gx  = idx % kW;
    const int gy  = y0 - kPad + p;        // global row
    xok[it]   = (gy >= 0) && (gy < kH);
    xgoff[it] = (uint32_t)((gy * kW + gx) * 4);
    xloff[it] = (uint32_t)((p * kXStr + gx + kPad) * 4);
  }

  auto issue_x = [&](int g, int buf) {
    const uint64_t xbase = (uint64_t)(uintptr_t)(
        x + (size_t)(nn * kIC + g * kWC + k) * kHW);
    const uint32_t ldsb = buf ? ldsX1 : ldsX0;
#pragma unroll
    for (int it = 0; it < 7; ++it) {
      if (xok[it]) {
        const uint32_t loff = ldsb + xloff[it];
        asm volatile("global_load_async_to_lds_b32 %0, %1, %2"
                     :: "v"(loff), "v"(xgoff[it]), "s"(xbase) : "memory");
      }
    }
  };

  // prologue: prefetch x for g=0; wait for TDM weight tile + x(0)
  issue_x(0, 0);
  asm volatile("s_wait_asynccnt 0x0" ::: "memory");
  asm volatile("s_wait_tensorcnt 0x0" ::: "memory");
  __syncthreads();

  const int wRowOff = rowIdx * kW + x0;   // offset within one tap's 8x96 LDS slab

  for (int g = 0; g < kG; ++g) {
    const int cur = g & 1;
    if (g + 1 < kG) issue_x(g + 1, cur ^ 1);   // overlap next channel's halo copy

    const float* Xb = &XT[cur][0];
    float a0 = 0.f, a1 = 0.f, a2 = 0.f, a3 = 0.f;
#pragma unroll
    for (int i = 0; i < kKS; ++i) {
      // 12-float sliding window for this tap-row: 3 aligned b128 LDS reads
      const float4* xr = (const float4*)(Xb + (rowIdx + i) * kXStr + x0);
      const float4 va = xr[0], vb = xr[1], vc = xr[2];
      const float xw[12] = {va.x, va.y, va.z, va.w,
                            vb.x, vb.y, vb.z, vb.w,
                            vc.x, vc.y, vc.z, vc.w};
#pragma unroll
      for (int j = 0; j < kKS; ++j) {
        const float4 wv =
            *(const float4*)(&WT[(i * kKS + j) * (kTH * kW) + wRowOff]);
        a0 = fmaf(xw[j + 0], wv.x, a0);
        a1 = fmaf(xw[j + 1], wv.y, a1);
        a2 = fmaf(xw[j + 2], wv.z, a2);
        a3 = fmaf(xw[j + 3], wv.w, a3);
      }
    }

    float4 r;
    r.x = a0; r.y = a1; r.z = a2; r.w = a3;
    float* op = out + (size_t)(nn * kIC + g * kWC + k) * kHW
                    + (size_t)(y0 + rowIdx) * kW + x0;
    *(float4*)op = r;                      // aligned b128 store

    // complete prefetch for g+1 and make buf[cur] safe to overwrite at g+2
    asm volatile("s_wait_asynccnt 0x0" ::: "memory");
    __syncthreads();
  }
}

extern "C" void kernel_launch(void* const* d_in, const int* /*in_sizes*/, int /*n_in*/,
                              void* d_out, int /*out_size*/, void* /*d_ws*/, size_t /*ws_size*/,
                              hipStream_t stream) {
  const float* x = (const float*)d_in[0];
  const float* w = (const float*)d_in[1];
  float* out = (float*)d_out;
  dim3 grid(kH / kTH, kN * kWC);   // (12, 64) = 768 blocks
  ska_forward_kernel<<<grid, kNT, 0, stream>>>(x, w, out);
}